// LinearSelfAttention_14413910245888
// MI455X (gfx1250) — compile-verified
//
#include <hip/hip_runtime.h>
#include <math.h>

// MI455X / gfx1250: wave32, WMMA f32 16x16x4 keeps fp32 numerics on the matrix pipe.
// GEMM staging uses the Tensor Data Mover (tensor_load_to_lds) with double buffering.
typedef __attribute__((ext_vector_type(2))) float v2f;
typedef __attribute__((ext_vector_type(8))) float v8f;
typedef unsigned u32x4 __attribute__((ext_vector_type(4)));
typedef unsigned u32x8 __attribute__((ext_vector_type(8)));

#define BATCH 2
#define SEQ 4096
#define DIMM 1024
#define NHEADS 16
#define DH 64
#define LHEADS 8
#define WIN 128
#define NWIN 32
#define MROWS (BATCH * SEQ)   // 8192

__device__ __forceinline__ float neg_max() { return -3.402823466e38f; }

// ---------------------------------------------------------------------------
// TDM: issue one tensor_load_to_lds moving a padded 2D tile (fp32 elements).
//   ldsByteOff      : byte offset of destination inside the wave's LDS alloc
//   gaddr           : global address of tile start
//   td0/td1         : tensor dims (elements)     stride0: dim0 line stride (elements)
//   tile0/tile1     : tile dims (elements)
//   padIntCode      : pad_interval code (4 -> every 32 DWORDs, 6 -> every 128 DWORDs)
//   padAmtCode      : pad_amount code   (3 -> 4 DWORDs,        7 -> 8 DWORDs)
// D# layout per CDNA5 ISA. Groups 2/3 zero (2D tensor).
// ---------------------------------------------------------------------------
__device__ __forceinline__ void tdm_load_2d(unsigned ldsByteOff, const void* gaddr,
                                            unsigned td0, unsigned td1, unsigned stride0,
                                            unsigned tile0, unsigned tile1,
                                            unsigned padIntCode, unsigned padAmtCode) {
    unsigned long long ga = (unsigned long long)gaddr;
    u32x4 g0;
    g0.x = 1u;                                                  // count=1, user mode
    g0.y = ldsByteOff;                                          // lds_addr
    g0.z = (unsigned)(ga & 0xFFFFFFFFu);                        // global_addr[31:0]
    g0.w = (unsigned)((ga >> 32) & 0x01FFFFFFu) | (2u << 30);   // addr[56:32] | type=2
    u32x8 g1;
    g1.s0 = (2u << 16) | (1u << 20) | (padIntCode << 22) | (padAmtCode << 25); // 4B elems, pad
    g1.s1 = (td0 & 0xFFFFu) << 16;                              // tensor_dim0[15:0]
    g1.s2 = ((td0 >> 16) & 0xFFFFu) | ((td1 & 0xFFFFu) << 16);  // td0 hi | td1 lo
    g1.s3 = ((td1 >> 16) & 0xFFFFu) | ((tile0 & 0xFFFFu) << 16);// td1 hi | tile_dim0
    g1.s4 = tile1 & 0xFFFFu;                                    // tile_dim1 (tile_dim2=0)
    g1.s5 = stride0;                                            // tensor_dim0_stride[31:0]
    g1.s6 = 0u;
    g1.s7 = 0u;
    u32x4 gz;
    gz.x = 0u; gz.y = 0u; gz.z = 0u; gz.w = 0u;
    asm volatile("tensor_load_to_lds %0, %1, %2, %3"
                 :: "s"(g0), "s"(g1), "s"(gz), "s"(gz)
                 : "memory");
}

// ---------------------------------------------------------------------------
// fp32 WMMA GEMM: C[M,N] = A[M,K] @ B[K,N] (+ bias). M%128==0, N%128==0, K%32==0.
// Block = 256 threads (8 waves). Block tile 128x128, wave tile 64x32 (4x2 frags).
// TDM double-buffered staging: As[2] 128x36, Bs[2] 32x136 (pads give conflict-free
// fragment reads: both lane-halves of a WMMA operand hit disjoint LDS banks).
// ---------------------------------------------------------------------------
#define LDS_A_FLOATS (128 * 36)
#define LDS_B_FLOATS (32 * 136)

__global__ __launch_bounds__(256)
void gemm_f32_wmma(const float* __restrict__ A, const float* __restrict__ B,
                   const float* __restrict__ bias, float* __restrict__ C,
                   int M, int N, int K) {
    extern __shared__ float smem[];
    float* As0 = smem;
    float* As1 = smem + LDS_A_FLOATS;
    float* Bs0 = smem + 2 * LDS_A_FLOATS;
    float* Bs1 = smem + 2 * LDS_A_FLOATS + LDS_B_FLOATS;
    // Dynamic-LDS base offset is 0 (no static LDS here) -> descriptor byte offsets:
    const unsigned offA0 = 0;
    const unsigned offA1 = LDS_A_FLOATS * 4;
    const unsigned offB0 = 2 * LDS_A_FLOATS * 4;
    const unsigned offB1 = (2 * LDS_A_FLOATS + LDS_B_FLOATS) * 4;

    const int tid  = threadIdx.x;
    const int lane = tid & 31;
    const int wave = tid >> 5;
    const int half = lane >> 4;
    const int l15  = lane & 15;
    const int mBlk = blockIdx.y * 128;
    const int nBlk = blockIdx.x * 128;
    const int wv_m = wave >> 2;   // 0..1
    const int wv_n = wave & 3;    // 0..3

    v8f acc[4][2];
    #pragma unroll
    for (int mt = 0; mt < 4; ++mt)
        #pragma unroll
        for (int nt = 0; nt < 2; ++nt)
            #pragma unroll
            for (int r = 0; r < 8; ++r) acc[mt][nt][r] = 0.0f;

    const int nChunks = K / 32;
    if (wave == 0) {
        tdm_load_2d(offA0, &A[(size_t)mBlk * K], K, M, K, 32, 128, 4, 3);
        tdm_load_2d(offB0, &B[nBlk],             N, K, N, 128, 32, 6, 7);
    }

    for (int c = 0; c < nChunks; ++c) {
        const int buf = c & 1;
        if (wave == 0) {
            if (c + 1 < nChunks) {
                const int ko = (c + 1) * 32;
                tdm_load_2d(buf ? offA0 : offA1, &A[(size_t)mBlk * K + ko], K, M, K, 32, 128, 4, 3);
                tdm_load_2d(buf ? offB0 : offB1, &B[(size_t)ko * N + nBlk], N, K, N, 128, 32, 6, 7);
                __builtin_amdgcn_s_wait_tensorcnt(2);   // current pair complete, next in flight
            } else {
                __builtin_amdgcn_s_wait_tensorcnt(0);
            }
        }
        __syncthreads();                                 // tile `buf` visible to all waves

        const float* Acur = buf ? As1 : As0;
        const float* Bcur = buf ? Bs1 : Bs0;
        #pragma unroll
        for (int kk = 0; kk < 32; kk += 4) {
            v2f af[4], bf[2];
            #pragma unroll
            for (int mt = 0; mt < 4; ++mt) {
                int m = wv_m * 64 + mt * 16 + l15;
                af[mt].x = Acur[m * 36 + kk + 2 * half];   // A 16x4: lanes0-15 K0/K1, 16-31 K2/K3
                af[mt].y = Acur[m * 36 + kk + 2 * half + 1];
            }
            #pragma unroll
            for (int nt = 0; nt < 2; ++nt) {
                int n = wv_n * 32 + nt * 16 + l15;
                bf[nt].x = Bcur[(kk + 2 * half) * 136 + n];
                bf[nt].y = Bcur[(kk + 2 * half + 1) * 136 + n];
            }
            #pragma unroll
            for (int mt = 0; mt < 4; ++mt)
                #pragma unroll
                for (int nt = 0; nt < 2; ++nt)
                    acc[mt][nt] = __builtin_amdgcn_wmma_f32_16x16x4_f32(
                        false, af[mt], false, bf[nt], (short)0, acc[mt][nt], false, false);
        }
        __syncthreads();                                 // compute done before buf is re-filled
    }

    #pragma unroll
    for (int mt = 0; mt < 4; ++mt)
        #pragma unroll
        for (int nt = 0; nt < 2; ++nt) {
            int n = nBlk + wv_n * 32 + nt * 16 + l15;
            float bv = bias ? bias[n] : 0.0f;
            #pragma unroll
            for (int r = 0; r < 8; ++r) {
                int m = mBlk + wv_m * 64 + mt * 16 + r + 8 * half;  // C/D: row r+8*half, col l15
                C[(size_t)m * N + n] = acc[mt][nt][r] + bv;
            }
        }
}

// ---------------------------------------------------------------------------
// Local windowed attention, heads 0..7. Block = (b, h, window, 64-query chunk).
// Dynamic LDS: Qs 64x68 | Ks 384x68 (reused for V) | Ss 64x388  = 216 KB (< 320 KB/WGP).
// ---------------------------------------------------------------------------
__global__ __launch_bounds__(256)
void local_attn_wmma(const float* __restrict__ Q, const float* __restrict__ Kt,
                     const float* __restrict__ V, const unsigned char* __restrict__ msk,
                     float* __restrict__ ATT) {
    extern __shared__ float smem[];
    float* Qs = smem;                 // [64][68]
    float* Ks = Qs + 64 * 68;         // [384][68], reused as Vs in phase 2
    float* Ss = Ks + 384 * 68;        // [64][388]

    const int tid  = threadIdx.x;
    const int lane = tid & 31;
    const int wave = tid >> 5;
    const int half = lane >> 4;
    const int l15  = lane & 15;

    int bid = blockIdx.x;
    const int bb = bid >> 9;               // 0..1
    const int h  = (bid >> 6) & 7;         // 0..7
    const int w  = (bid >> 1) & 31;        // window 0..31
    const int qc = bid & 1;                // 64-query chunk
    const int tq0 = w * WIN + qc * 64;     // first query t in chunk
    const size_t hcol = (size_t)h * DH;
    const size_t bbase = (size_t)bb * SEQ;

    // Load Q chunk (64 x 64)
    #pragma unroll
    for (int it = 0; it < 4; ++it) {
        int idx = it * 256 + tid;
        int r = idx >> 4;
        int c = (idx & 15) * 4;
        float4 v = *reinterpret_cast<const float4*>(&Q[(bbase + tq0 + r) * DIMM + hcol + c]);
        *reinterpret_cast<float4*>(&Qs[r * 68 + c]) = v;
    }
    // Load K look-around (384 x 64), zero for OOB windows
    #pragma unroll
    for (int it = 0; it < 24; ++it) {
        int idx = it * 256 + tid;
        int r = idx >> 4;                  // key row 0..383
        int c = (idx & 15) * 4;
        int wk = w - 1 + (r >> 7);
        float4 v = make_float4(0.f, 0.f, 0.f, 0.f);
        if ((unsigned)wk < (unsigned)NWIN)
            v = *reinterpret_cast<const float4*>(&Kt[(bbase + wk * WIN + (r & 127)) * DIMM + hcol + c]);
        *reinterpret_cast<float4*>(&Ks[r * 68 + c]) = v;
    }
    __syncthreads();

    // Phase 1: S = Q @ K^T (64 x 384), scale + mask, into Ss.
    {
        const int wv_m = wave >> 1;        // 0..3 -> 16-row tile
        const int wv_n = wave & 1;         // 0..1 -> 12 N-tiles each
        v8f acc[12];
        #pragma unroll
        for (int nt = 0; nt < 12; ++nt)
            #pragma unroll
            for (int r = 0; r < 8; ++r) acc[nt][r] = 0.0f;

        for (int kk = 0; kk < DH; kk += 4) {
            v2f a;
            int m = wv_m * 16 + l15;
            a.x = Qs[m * 68 + kk + 2 * half];
            a.y = Qs[m * 68 + kk + 2 * half + 1];
            #pragma unroll
            for (int nt = 0; nt < 12; ++nt) {
                int key = (wv_n * 12 + nt) * 16 + l15;
                v2f b;
                b.x = Ks[key * 68 + kk + 2 * half];      // B[kdim][key] = K[key][kdim]
                b.y = Ks[key * 68 + kk + 2 * half + 1];
                acc[nt] = __builtin_amdgcn_wmma_f32_16x16x4_f32(
                    false, a, false, b, (short)0, acc[nt], false, false);
            }
        }
        #pragma unroll
        for (int nt = 0; nt < 12; ++nt) {
            int kidx = (wv_n * 12 + nt) * 16 + l15;
            int wk = w - 1 + (kidx >> 7);
            bool kok = ((unsigned)wk < (unsigned)NWIN);
            int tk = wk * WIN + (kidx & 127);
            bool kval = kok && (msk[bbase + tk] == 0);
            #pragma unroll
            for (int r = 0; r < 8; ++r) {
                int mq = wv_m * 16 + r + 8 * half;
                bool qval = (msk[bbase + tq0 + mq] == 0);
                float s = (kval && qval) ? acc[nt][r] * 0.125f : neg_max();
                Ss[mq * 388 + kidx] = s;
            }
        }
    }
    __syncthreads();

    // Load V look-around into Ks region (zero-padded OOB) + softmax rows of Ss.
    #pragma unroll
    for (int it = 0; it < 24; ++it) {
        int idx = it * 256 + tid;
        int r = idx >> 4;
        int c = (idx & 15) * 4;
        int wk = w - 1 + (r >> 7);
        float4 v = make_float4(0.f, 0.f, 0.f, 0.f);
        if ((unsigned)wk < (unsigned)NWIN)
            v = *reinterpret_cast<const float4*>(&V[(bbase + wk * WIN + (r & 127)) * DIMM + hcol + c]);
        *reinterpret_cast<float4*>(&Ks[r * 68 + c]) = v;
    }
    {
        #pragma unroll
        for (int rr = 0; rr < 8; ++rr) {
            int m = wave * 8 + rr;
            float mx = neg_max();
            float sv[12];
            #pragma unroll
            for (int j = 0; j < 12; ++j) {
                sv[j] = Ss[m * 388 + lane + j * 32];
                mx = fmaxf(mx, sv[j]);
            }
            #pragma unroll
            for (int off = 16; off >= 1; off >>= 1) mx = fmaxf(mx, __shfl_xor(mx, off));
            float sum = 0.f;
            #pragma unroll
            for (int j = 0; j < 12; ++j) { sv[j] = expf(sv[j] - mx); sum += sv[j]; }
            #pragma unroll
            for (int off = 16; off >= 1; off >>= 1) sum += __shfl_xor(sum, off);
            float inv = 1.0f / sum;
            #pragma unroll
            for (int j = 0; j < 12; ++j) Ss[m * 388 + lane + j * 32] = sv[j] * inv;
        }
    }
    __syncthreads();

    // Phase 2: O = P(64x384) @ V(384x64)
    {
        const int wv_m = wave >> 1;        // 0..3
        const int wv_n = wave & 1;         // 0..1 -> 2 N-tiles each
        v8f acc[2];
        #pragma unroll
        for (int nt = 0; nt < 2; ++nt)
            #pragma unroll
            for (int r = 0; r < 8; ++r) acc[nt][r] = 0.0f;

        for (int kk = 0; kk < 384; kk += 4) {
            v2f a;
            int m = wv_m * 16 + l15;
            a.x = Ss[m * 388 + kk + 2 * half];
            a.y = Ss[m * 388 + kk + 2 * half + 1];
            #pragma unroll
            for (int nt = 0; nt < 2; ++nt) {
                int n = (wv_n * 2 + nt) * 16 + l15;
                v2f b;
                b.x = Ks[(kk + 2 * half) * 68 + n];
                b.y = Ks[(kk + 2 * half + 1) * 68 + n];
                acc[nt] = __builtin_amdgcn_wmma_f32_16x16x4_f32(
                    false, a, false, b, (short)0, acc[nt], false, false);
            }
        }
        #pragma unroll
        for (int nt = 0; nt < 2; ++nt) {
            int n = (wv_n * 2 + nt) * 16 + l15;
            #pragma unroll
            for (int r = 0; r < 8; ++r) {
                int mq = wv_m * 16 + r + 8 * half;
                ATT[(bbase + tq0 + mq) * DIMM + hcol + n] = acc[nt][r];
            }
        }
    }
}

// ---------------------------------------------------------------------------
// Linear (global) attention, heads 8..15. One block per (b, head).
// ---------------------------------------------------------------------------
__global__ __launch_bounds__(256)
void linear_attn_kernel(const float* __restrict__ Q, const float* __restrict__ Kt,
                        const float* __restrict__ V, const unsigned char* __restrict__ msk,
                        float* __restrict__ ATT) {
    __shared__ float ctx[64][64];
    __shared__ float red[64][4];
    __shared__ float kmax[64];
    __shared__ float ksum[64];
    __shared__ float Kc[32][64];
    __shared__ float Vc[32][64];

    const int tid = threadIdx.x;
    const int bb = blockIdx.x >> 3;
    const int h  = LHEADS + (blockIdx.x & 7);
    const size_t base = ((size_t)bb * SEQ) * DIMM + (size_t)h * DH;
    const size_t mbase = (size_t)bb * SEQ;

    const int d = tid & 63;
    const int g = tid >> 6;     // 0..3

    // Pass 1: column max of masked K over t
    float mx = neg_max();
    for (int t = g * 1024; t < (g + 1) * 1024; ++t) {
        float kv = Kt[base + (size_t)t * DIMM + d];
        float km = (msk[mbase + t] == 0) ? kv : neg_max();
        mx = fmaxf(mx, km);
    }
    red[d][g] = mx;
    __syncthreads();
    if (g == 0)
        kmax[d] = fmaxf(fmaxf(red[d][0], red[d][1]), fmaxf(red[d][2], red[d][3]));
    __syncthreads();

    // Pass 2: column sum of exp
    float sm = 0.f;
    for (int t = g * 1024; t < (g + 1) * 1024; ++t) {
        float kv = Kt[base + (size_t)t * DIMM + d];
        float km = (msk[mbase + t] == 0) ? kv : neg_max();
        sm += expf(km - kmax[d]);
    }
    red[d][g] = sm;
    __syncthreads();
    if (g == 0)
        ksum[d] = red[d][0] + red[d][1] + red[d][2] + red[d][3];
    __syncthreads();

    // Context = sum_t softmax_k[t,d] * v_masked[t,e]
    float acc[16];
    #pragma unroll
    for (int i = 0; i < 16; ++i) acc[i] = 0.f;
    const int e = tid & 63;
    const int dg = tid >> 6;

    for (int c = 0; c < SEQ / 32; ++c) {
        #pragma unroll
        for (int j = 0; j < 8; ++j) {
            int idx = j * 256 + tid;
            int tt = idx >> 6;
            int dd = idx & 63;
            int t = c * 32 + tt;
            bool val = (msk[mbase + t] == 0);
            float kv = Kt[base + (size_t)t * DIMM + dd];
            float km = val ? kv : neg_max();
            Kc[tt][dd] = expf(km - kmax[dd]) / ksum[dd];
            float vv = V[base + (size_t)t * DIMM + dd];
            Vc[tt][dd] = val ? vv : 0.f;
        }
        __syncthreads();
        #pragma unroll 4
        for (int tt = 0; tt < 32; ++tt) {
            float vv = Vc[tt][e];
            #pragma unroll
            for (int i = 0; i < 16; ++i) acc[i] += Kc[tt][dg * 16 + i] * vv;
        }
        __syncthreads();
    }
    #pragma unroll
    for (int i = 0; i < 16; ++i) ctx[dg * 16 + i][e] = acc[i];
    __syncthreads();

    // out = (softmax(q,-1) * d^-0.5) @ ctx ; one row per wave iteration
    const int lane = tid & 31;
    const int wave = tid >> 5;
    for (int r = wave; r < SEQ; r += 8) {
        size_t off = base + (size_t)r * DIMM;
        float q0 = Q[off + lane];
        float q1 = Q[off + lane + 32];
        float m2 = fmaxf(q0, q1);
        #pragma unroll
        for (int o = 16; o >= 1; o >>= 1) m2 = fmaxf(m2, __shfl_xor(m2, o));
        float e0 = expf(q0 - m2), e1 = expf(q1 - m2);
        float s = e0 + e1;
        #pragma unroll
        for (int o = 16; o >= 1; o >>= 1) s += __shfl_xor(s, o);
        float inv = 0.125f / s;
        float p0 = e0 * inv, p1 = e1 * inv;
        float o0 = 0.f, o1 = 0.f;
        #pragma unroll 8
        for (int dd = 0; dd < 32; ++dd) {
            float pd = __shfl(p0, dd);
            o0 += pd * ctx[dd][lane];
            o1 += pd * ctx[dd][lane + 32];
        }
        #pragma unroll 8
        for (int dd = 0; dd < 32; ++dd) {
            float pd = __shfl(p1, dd);
            o0 += pd * ctx[dd + 32][lane];
            o1 += pd * ctx[dd + 32][lane + 32];
        }
        ATT[off + lane] = o0;
        ATT[off + lane + 32] = o1;
    }
}

// ---------------------------------------------------------------------------
extern "C" void kernel_launch(void* const* d_in, const int* in_sizes, int n_in,
                              void* d_out, int out_size, void* d_ws, size_t ws_size,
                              hipStream_t stream) {
    const float* x = (const float*)d_in[0];
    const unsigned char* msk = (const unsigned char*)d_in[1];  // jax bool -> 1 byte
    const float* Wq = (const float*)d_in[2];
    const float* Wk = (const float*)d_in[3];
    const float* Wv = (const float*)d_in[4];
    const float* Wo = (const float*)d_in[5];
    const float* bo = (const float*)d_in[6];
    float* out = (float*)d_out;

    const size_t plane = (size_t)MROWS * DIMM;   // 8.39M floats
    float* Qb  = (float*)d_ws;
    float* Kb  = Qb + plane;
    float* Vb  = Kb + plane;
    float* ATT = Vb + plane;

    const size_t gemmLds = (size_t)(2 * LDS_A_FLOATS + 2 * LDS_B_FLOATS) * sizeof(float); // 71680
    dim3 gg(DIMM / 128, MROWS / 128);            // (8, 64)
    gemm_f32_wmma<<<gg, 256, gemmLds, stream>>>(x, Wq, nullptr, Qb, MROWS, DIMM, DIMM);
    gemm_f32_wmma<<<gg, 256, gemmLds, stream>>>(x, Wk, nullptr, Kb, MROWS, DIMM, DIMM);
    gemm_f32_wmma<<<gg, 256, gemmLds, stream>>>(x, Wv, nullptr, Vb, MROWS, DIMM, DIMM);

    size_t lds = (size_t)(64 * 68 + 384 * 68 + 64 * 388) * sizeof(float);  // 216 KB
    local_attn_wmma<<<BATCH * LHEADS * NWIN * 2, 256, lds, stream>>>(Qb, Kb, Vb, msk, ATT);
    linear_attn_kernel<<<BATCH * (NHEADS - LHEADS), 256, 0, stream>>>(Qb, Kb, Vb, msk, ATT);

    gemm_f32_wmma<<<gg, 256, gemmLds, stream>>>(ATT, Wo, bo, out, MROWS, DIMM, DIMM);
}